// RNN_90537910600159
// MI455X (gfx1250) — compile-verified
//
#include <hip/hip_runtime.h>

typedef float v2f __attribute__((ext_vector_type(2)));
typedef float v8f __attribute__((ext_vector_type(8)));

union V8 { v8f v; float f[8]; };

__device__ __forceinline__ float fast_tanh(float x) {
  // tanh(x) = (e^{2x} - 1) / (e^{2x} + 1), e^{2x} = 2^(2*log2(e)*x)
  float e = __builtin_amdgcn_exp2f(x * 2.8853900817779268f);
  return (e - 1.0f) * __builtin_amdgcn_rcpf(e + 1.0f);
}

// One wave32 owns a 16-row batch tile for the full T=512 sequence.
// Hidden state is kept transposed (h^T : M=hidden x N=batch) as a WMMA C/D
// fragment. K-permutation trick: hidden index k = 8*(K_pos>=2) + 2c + (K_pos&1)
// makes the per-chunk B operands pure register aliases of the D fragment
// (each lane half already owns exactly the 8 rows its B slots need), so the
// D -> next-B conversion costs ZERO cross-lane ops; the permutation is folded
// into the loop-invariant A operands instead.
__global__ __launch_bounds__(256) void rnn_fused_wmma(
    const float* __restrict__ x,      // [B,T,I]
    const float* __restrict__ W_ih,   // [H,I]
    const float* __restrict__ W_hh,   // [H,H]
    const float* __restrict__ b_ih,   // [H]
    const float* __restrict__ b_hh,   // [H]
    const float* __restrict__ W_out,  // [O,H]
    const float* __restrict__ b_out,  // [O]
    float* __restrict__ out)          // [B,T,O]
{
  constexpr int T = 512, I = 3, H = 16, O = 3;
  const int lane = threadIdx.x & 31;
  const int wv   = (blockIdx.x * (blockDim.x >> 5)) + (threadIdx.x >> 5);
  const int base = wv << 4;          // first batch row of this tile
  const int n    = lane & 15;        // N index (batch col) / A-matrix M row
  const int hl   = lane >> 4;        // lane half

  // ---- loop-invariant A operands (K-permuted) ---------------------------
  // A slot (VGPR r', half h') <-> K_pos = r' + 2h'; we map K_pos to hidden
  // index k = 8*h' + 2c + r', matching B_c = (hf[2c], hf[2c+1]).
  v2f A_hh[4], A_out[4];
#pragma unroll
  for (int c = 0; c < 4; ++c) {
    const int k0 = 8 * hl + 2 * c;
    A_hh[c].x = W_hh[n * H + k0];
    A_hh[c].y = W_hh[n * H + k0 + 1];
    A_out[c].x = (n < O) ? W_out[n * H + k0]     : 0.0f;  // W_out zero-padded to 16 rows
    A_out[c].y = (n < O) ? W_out[n * H + k0 + 1] : 0.0f;
  }

  // ---- per-lane input-projection weights + fused biases (C/D layout) ----
  // C/D fragment: VGPR r <-> hidden row q = r + 8*half, lane%16 = batch col.
  float w0[8], w1[8], w2[8], cb[8], cbo[8];
#pragma unroll
  for (int r = 0; r < 8; ++r) {
    const int q = r + 8 * hl;
    w0[r]  = W_ih[q * I + 0];
    w1[r]  = W_ih[q * I + 1];
    w2[r]  = W_ih[q * I + 2];
    cb[r]  = b_ih[q] + b_hh[q];
    cbo[r] = (hl == 0 && r < O) ? b_out[r] : 0.0f;
  }

  const float* xr  = x   + (size_t)(base + n) * (T * I);
  float*       orr = out + (size_t)(base + n) * (T * O);

  // hidden state h^T, stored directly as the 4 B-chunk operands
  v2f Hp[4];
#pragma unroll
  for (int c = 0; c < 4; ++c) { Hp[c].x = 0.0f; Hp[c].y = 0.0f; }

  for (int t = 0; t < T; ++t) {
    // stream x[b, t, 0..2] (12B per lane; lanes 16-31 hit same lines in L0)
    const float x0 = xr[t * I + 0];
    const float x1 = xr[t * I + 1];
    const float x2 = xr[t * I + 2];
    if (t + 16 < T) __builtin_prefetch(xr + (t + 16) * I, 0, 0);

    // C = x @ W_ih^T + b_ih + b_hh, directly in C/D fragment layout
    V8 acc;
#pragma unroll
    for (int r = 0; r < 8; ++r)
      acc.f[r] = fmaf(x0, w0[r], fmaf(x1, w1[r], fmaf(x2, w2[r], cb[r])));

    // pre^T = W_hh @ h^T + C  (4 chained K=4 WMMAs; B == Hp, no shuffles)
#pragma unroll
    for (int c = 0; c < 4; ++c)
      acc.v = __builtin_amdgcn_wmma_f32_16x16x4_f32(
          false, A_hh[c], false, Hp[c], (short)0, acc.v, false, false);

    // h = tanh(pre); D VGPR r holds row 8*hl + r, and B chunk c needs rows
    // (8*hl + 2c, 8*hl + 2c+1) -> Hp[c] = tanh of acc pair (2c, 2c+1).
#pragma unroll
    for (int c = 0; c < 4; ++c) {
      Hp[c].x = fast_tanh(acc.f[2 * c + 0]);
      Hp[c].y = fast_tanh(acc.f[2 * c + 1]);
    }

    // out^T = W_out_pad @ h^T + b_out  (reuses the same B operands)
    V8 oa;
#pragma unroll
    for (int r = 0; r < 8; ++r) oa.f[r] = cbo[r];
#pragma unroll
    for (int c = 0; c < 4; ++c)
      oa.v = __builtin_amdgcn_wmma_f32_16x16x4_f32(
          false, A_out[c], false, Hp[c], (short)0, oa.v, false, false);

    // rows 0..2 of out^T live in VGPRs 0..2 of lanes 0-15
    if (hl == 0) {
      orr[t * O + 0] = oa.f[0];
      orr[t * O + 1] = oa.f[1];
      orr[t * O + 2] = oa.f[2];
    }
  }
}

extern "C" void kernel_launch(void* const* d_in, const int* in_sizes, int n_in,
                              void* d_out, int out_size, void* d_ws, size_t ws_size,
                              hipStream_t stream) {
  (void)in_sizes; (void)n_in; (void)d_ws; (void)ws_size; (void)out_size;
  const float* x     = (const float*)d_in[0];
  const float* W_ih  = (const float*)d_in[1];
  const float* W_hh  = (const float*)d_in[2];
  const float* b_ih  = (const float*)d_in[3];
  const float* b_hh  = (const float*)d_in[4];
  const float* W_out = (const float*)d_in[5];
  const float* b_out = (const float*)d_in[6];
  float* out = (float*)d_out;

  const int tiles = 4096 / 16;      // 256 wave-tiles
  const int wavesPerBlock = 8;      // 256 threads = 8 wave32
  dim3 block(32 * wavesPerBlock);
  dim3 grid(tiles / wavesPerBlock); // 32 blocks
  hipLaunchKernelGGL(rnn_fused_wmma, grid, block, 0, stream,
                     x, W_ih, W_hh, b_ih, b_hh, W_out, b_out, out);
}